// GraphConv_52819507806605
// MI455X (gfx1250) — compile-verified
//
#include <hip/hip_runtime.h>
#include <math.h>

// ---------------------------------------------------------------------------
// Problem constants (from setup_inputs)
// ---------------------------------------------------------------------------
#define N_NODES   8192
#define N_EDGES   262144
#define DE_DIM    64
#define KNN_K     16
#define EDGES_DYN (N_NODES * (KNN_K + 1))   // 139264, multiple of 256
#define BN_RPC    2048                       // BN rows per partial chunk

typedef __attribute__((ext_vector_type(2))) float v2f;
typedef __attribute__((ext_vector_type(8))) float v8f;

// ---------------------------------------------------------------------------
// WMMA fp32 GEMM: C[M,N] = A[M,K] @ B[K,N], row-major with strides.
// Each wave32 computes a 64x32 tile as 4x2 V_WMMA_F32_16X16X4_F32 ops:
// A fragments reused x2, B fragments reused x4 (~10.7 FLOP/B vs 4 for 1x1).
// Block = 4 waves along M -> 256x32 block tile (all M are multiples of 256).
// A frag (16x4): lane<16 -> K={k0,k0+1}, lane>=16 -> K={k0+2,k0+3}.
// B frag (4x16): VGPR0 rows {k0,k0+2}, VGPR1 rows {k0+1,k0+3} per half.
// C/D (16x16): VGPR r -> row r (lanes 0-15) / row r+8 (lanes 16-31).
// ---------------------------------------------------------------------------
__global__ __launch_bounds__(128) void gemm_f32_wmma(
    const float* __restrict__ A, const float* __restrict__ B,
    float* __restrict__ C, int M, int K, int N, int lda, int ldb, int ldc)
{
    const int lane = threadIdx.x & 31;
    const int wave = threadIdx.x >> 5;
    const int n0 = blockIdx.x * 32;
    const int m0 = (blockIdx.y * 4 + wave) * 64;
    if (m0 >= M) return;
    const int half = lane >> 4;      // 0 or 1
    const int l16  = lane & 15;

    v8f acc[4][2];
#pragma unroll
    for (int mi = 0; mi < 4; ++mi)
#pragma unroll
        for (int ni = 0; ni < 2; ++ni)
            acc[mi][ni] = (v8f){0.f, 0.f, 0.f, 0.f, 0.f, 0.f, 0.f, 0.f};

    const float* arow[4];
#pragma unroll
    for (int mi = 0; mi < 4; ++mi)
        arow[mi] = A + (size_t)(m0 + mi * 16 + l16) * lda;

    for (int k0 = 0; k0 < K; k0 += 4) {
        const int ka = k0 + half * 2;
        const float* brow0 = B + (size_t)ka * ldb + n0 + l16;
        const float* brow1 = B + (size_t)(ka + 1) * ldb + n0 + l16;
        v2f a[4], b[2];
#pragma unroll
        for (int mi = 0; mi < 4; ++mi) {
            a[mi].x = arow[mi][ka];
            a[mi].y = arow[mi][ka + 1];
        }
        b[0].x = brow0[0];   b[0].y = brow1[0];
        b[1].x = brow0[16];  b[1].y = brow1[16];
#pragma unroll
        for (int mi = 0; mi < 4; ++mi) {
#pragma unroll
            for (int ni = 0; ni < 2; ++ni) {
                acc[mi][ni] = __builtin_amdgcn_wmma_f32_16x16x4_f32(
                    false, a[mi], false, b[ni], (short)0, acc[mi][ni], false, false);
            }
        }
    }
#pragma unroll
    for (int mi = 0; mi < 4; ++mi) {
#pragma unroll
        for (int r = 0; r < 8; ++r) {
            const int row = m0 + mi * 16 + half * 8 + r;
            C[(size_t)row * ldc + n0 + l16]      = acc[mi][0][r];
            C[(size_t)row * ldc + n0 + 16 + l16] = acc[mi][1][r];
        }
    }
}

// ---------------------------------------------------------------------------
// BatchNorm (training batch stats), deterministic two-stage reduction with
// coalesced loads. Stage 1: 32-column x BN_RPC-row tiles -> partial sums.
// Stage 2: fold partials per column -> mean, rsqrt(var + 1e-5).
// ---------------------------------------------------------------------------
__global__ __launch_bounds__(256) void bn_partial(
    const float* __restrict__ X, int M, int C, int ld,
    float* __restrict__ psum, float* __restrict__ psum2)
{
    const int c  = blockIdx.x * 32 + threadIdx.x;   // gridDim.x = C/32
    const int ch = blockIdx.y;
    const int r0 = ch * BN_RPC;
    int r1 = r0 + BN_RPC; if (r1 > M) r1 = M;

    float s = 0.f, s2 = 0.f;
    for (int r = r0 + threadIdx.y; r < r1; r += 8) {
        float v = X[(size_t)r * ld + c];
        s += v; s2 += v * v;
    }
    __shared__ float sh[8][32], sh2[8][32];
    sh[threadIdx.y][threadIdx.x]  = s;
    sh2[threadIdx.y][threadIdx.x] = s2;
    __syncthreads();
    if (threadIdx.y == 0) {
        float ts = 0.f, ts2 = 0.f;
#pragma unroll
        for (int t = 0; t < 8; ++t) { ts += sh[t][threadIdx.x]; ts2 += sh2[t][threadIdx.x]; }
        psum[(size_t)ch * C + c]  = ts;
        psum2[(size_t)ch * C + c] = ts2;
    }
}

__global__ __launch_bounds__(256) void bn_finalize(
    const float* __restrict__ psum, const float* __restrict__ psum2,
    int chunks, int C, int M, float* __restrict__ mean, float* __restrict__ rstd)
{
    const int c = blockIdx.x * blockDim.x + threadIdx.x;
    if (c >= C) return;
    float s = 0.f, s2 = 0.f;
    for (int ch = 0; ch < chunks; ++ch) {
        s  += psum[(size_t)ch * C + c];
        s2 += psum2[(size_t)ch * C + c];
    }
    float m = s / (float)M;
    mean[c] = m;
    rstd[c] = rsqrtf(s2 / (float)M - m * m + 1e-5f);
}

__global__ __launch_bounds__(256) void bn_lrelu(
    float* __restrict__ X, const float* __restrict__ mean,
    const float* __restrict__ rstd, float slope, int M, int C, int ld)
{
    size_t total = (size_t)M * C;
    size_t stride = (size_t)gridDim.x * blockDim.x;
    for (size_t i = (size_t)blockIdx.x * blockDim.x + threadIdx.x; i < total; i += stride) {
        int r = (int)(i / C), c = (int)(i % C);
        float v = (X[(size_t)r * ld + c] - mean[c]) * rstd[c];
        X[(size_t)r * ld + c] = (v >= 0.f) ? v : slope * v;
    }
}

// ---------------------------------------------------------------------------
// Static edge-conv scatter-max with order-preserving u32 encoding
// msg = [x_dst, x_src - x_dst, edge_attr]; agg = segment_max over dst.
// u32-max atomics are order-independent -> bit-deterministic across replays.
// ---------------------------------------------------------------------------
__device__ __forceinline__ unsigned enc_f32(float f) {
    unsigned u = __float_as_uint(f);
    return (u & 0x80000000u) ? ~u : (u | 0x80000000u);
}
__device__ __forceinline__ float dec_f32(unsigned u) {
    return (u & 0x80000000u) ? __uint_as_float(u ^ 0x80000000u)
                             : __uint_as_float(~u);
}

__global__ __launch_bounds__(256) void static_scatter(
    const float* __restrict__ x, const int* __restrict__ ei,
    const float* __restrict__ ea, unsigned* __restrict__ agg,
    int d)
{
    const int Cw = 2 * d + DE_DIM;
    size_t total = (size_t)N_EDGES * Cw;
    size_t stride = (size_t)gridDim.x * blockDim.x;
    for (size_t i = (size_t)blockIdx.x * blockDim.x + threadIdx.x; i < total; i += stride) {
        int e = (int)(i / Cw), c = (int)(i % Cw);
        int s   = ei[e];
        int dst = ei[N_EDGES + e];
        float val;
        if (c < d)            val = x[(size_t)dst * d + c];
        else if (c < 2 * d)   val = x[(size_t)s * d + (c - d)] - x[(size_t)dst * d + (c - d)];
        else                  val = ea[(size_t)e * DE_DIM + (c - 2 * d)];
        atomicMax(&agg[(size_t)dst * Cw + c], enc_f32(val));
    }
}

__global__ __launch_bounds__(256) void decode_agg(unsigned* __restrict__ agg, size_t total)
{
    size_t stride = (size_t)gridDim.x * blockDim.x;
    for (size_t i = (size_t)blockIdx.x * blockDim.x + threadIdx.x; i < total; i += stride) {
        unsigned u = agg[i];
        ((float*)agg)[i] = (u == 0u) ? 0.f : dec_f32(u);   // no in-edge -> 0
    }
}

// ---------------------------------------------------------------------------
// Row L2-normalize (for cosine kNN)
// ---------------------------------------------------------------------------
__global__ __launch_bounds__(128) void rownorm(
    const float* __restrict__ x, float* __restrict__ xn, int d)
{
    const int row = blockIdx.x;
    __shared__ float sh[128];
    float s = 0.f;
    for (int c = threadIdx.x; c < d; c += blockDim.x) {
        float v = x[(size_t)row * d + c];
        s += v * v;
    }
    sh[threadIdx.x] = s;
    __syncthreads();
    for (int o = 64; o > 0; o >>= 1) {
        if (threadIdx.x < o) sh[threadIdx.x] += sh[threadIdx.x + o];
        __syncthreads();
    }
    float inv = 1.f / (sqrtf(sh[0]) + 1e-12f);
    for (int c = threadIdx.x; c < d; c += blockDim.x)
        xn[(size_t)row * d + c] = x[(size_t)row * d + c] * inv;
}

// ---------------------------------------------------------------------------
// kNN (cosine): one wave32 per node; per-lane top-16 in registers, merged
// through LDS with shfl reductions. Tie-break = lowest index (jax top_k).
// Neighbor rows read as float4 (rows are 256B-aligned, d % 4 == 0).
// ---------------------------------------------------------------------------
__global__ __launch_bounds__(128) void knn_kernel(
    const float* __restrict__ xn, int* __restrict__ knn, int d)
{
    __shared__ __align__(16) float xi[4][128];
    __shared__ float mv[4][32 * 16];
    __shared__ int   mi[4][32 * 16];
    const int wave = threadIdx.x >> 5;
    const int lane = threadIdx.x & 31;
    const int i = blockIdx.x * 4 + wave;
    const bool active = (i < N_NODES);

    if (active)
        for (int c = lane; c < d; c += 32) xi[wave][c] = xn[(size_t)i * d + c];
    __syncthreads();
    if (!active) return;

    float vals[16]; int idxs[16];
#pragma unroll
    for (int t = 0; t < 16; ++t) { vals[t] = -INFINITY; idxs[t] = 0x7fffffff; }

    const int d4 = d >> 2;
    const float4* xi4 = (const float4*)&xi[wave][0];
    for (int j = lane; j < N_NODES; j += 32) {
        if (j == i) continue;
        const float4* xj4 = (const float4*)(xn + (size_t)j * d);
        float s = 0.f;
        for (int c = 0; c < d4; ++c) {
            float4 a = xi4[c], b = xj4[c];
            s += a.x * b.x + a.y * b.y + a.z * b.z + a.w * b.w;
        }
        if (s > vals[0]) {                // vals kept ascending; vals[0] = min
            int p = 0;
            while (p < 15 && s > vals[p + 1]) {
                vals[p] = vals[p + 1]; idxs[p] = idxs[p + 1]; ++p;
            }
            vals[p] = s; idxs[p] = j;
        }
    }

#pragma unroll
    for (int t = 0; t < 16; ++t) {
        mv[wave][lane * 16 + t] = vals[t];
        mi[wave][lane * 16 + t] = idxs[t];
    }
    // each lane only reads/writes its own LDS slots below; cross-lane via shfl
    for (int r = 0; r < KNN_K; ++r) {
        float bv = -INFINITY; int bi = 0x7fffffff; int bp = -1;
        for (int t = 0; t < 16; ++t) {
            float v = mv[wave][lane * 16 + t];
            int  id = mi[wave][lane * 16 + t];
            if (v > bv || (v == bv && id < bi)) { bv = v; bi = id; bp = t; }
        }
        for (int off = 16; off > 0; off >>= 1) {
            float ov = __shfl_xor(bv, off, 32);
            int   oi = __shfl_xor(bi, off, 32);
            if (ov > bv || (ov == bv && oi < bi)) { bv = ov; bi = oi; }
        }
        if (bp >= 0 && mv[wave][lane * 16 + bp] == bv && mi[wave][lane * 16 + bp] == bi)
            mv[wave][lane * 16 + bp] = -INFINITY;   // indices unique per lane
        if (lane == 0) knn[(size_t)i * KNN_K + r] = bi;
    }
}

// ---------------------------------------------------------------------------
// Dynamic-conv edge message build: edge e = i*17 + t; t<16 -> src=knn, t=16 self
// msg = [x_i, x_src - x_i]  -> [EDGES_DYN, 2d], grouped contiguously by dst
// ---------------------------------------------------------------------------
__global__ __launch_bounds__(256) void dyn_msg(
    const float* __restrict__ x, const int* __restrict__ knn,
    float* __restrict__ msg, int d)
{
    const int C = 2 * d;
    size_t total = (size_t)EDGES_DYN * C;
    size_t stride = (size_t)gridDim.x * blockDim.x;
    for (size_t idx = (size_t)blockIdx.x * blockDim.x + threadIdx.x; idx < total; idx += stride) {
        size_t e = idx / C; int c = (int)(idx % C);
        int i = (int)(e / (KNN_K + 1)), t = (int)(e % (KNN_K + 1));
        int s = (t < KNN_K) ? knn[(size_t)i * KNN_K + t] : i;
        float v;
        if (c < d) v = x[(size_t)i * d + c];
        else       v = x[(size_t)s * d + (c - d)] - x[(size_t)i * d + (c - d)];
        msg[idx] = v;
    }
}

__global__ __launch_bounds__(256) void dyn_segmax(
    const float* __restrict__ h, float* __restrict__ out, int C)
{
    size_t total = (size_t)N_NODES * C;
    size_t stride = (size_t)gridDim.x * blockDim.x;
    for (size_t idx = (size_t)blockIdx.x * blockDim.x + threadIdx.x; idx < total; idx += stride) {
        int i = (int)(idx / C), c = (int)(idx % C);
        float m = -INFINITY;
        for (int t = 0; t < KNN_K + 1; ++t) {
            float v = h[((size_t)i * (KNN_K + 1) + t) * C + c];
            m = (v > m) ? v : m;
        }
        out[idx] = m;   // self-loop guarantees finite
    }
}

__global__ __launch_bounds__(256) void copy_cols(
    const float* __restrict__ src, int srcC, float* __restrict__ dst, int ldDst)
{
    size_t total = (size_t)N_NODES * srcC;
    size_t stride = (size_t)gridDim.x * blockDim.x;
    for (size_t idx = (size_t)blockIdx.x * blockDim.x + threadIdx.x; idx < total; idx += stride) {
        int r = (int)(idx / srcC), c = (int)(idx % srcC);
        dst[(size_t)r * ldDst + c] = src[idx];
    }
}

// ---------------------------------------------------------------------------
// Host orchestration
// ---------------------------------------------------------------------------
static inline int ew_blocks(size_t total) {
    size_t b = (total + 255) / 256;
    return (int)(b > 4096 ? 4096 : b);
}

extern "C" void kernel_launch(void* const* d_in, const int* in_sizes, int n_in,
                              void* d_out, int out_size, void* d_ws, size_t ws_size,
                              hipStream_t stream) {
    (void)in_sizes; (void)n_in; (void)out_size; (void)ws_size;
    const int N = N_NODES;

    const float* x      = (const float*)d_in[0];
    const int*   ei     = (const int*)d_in[1];
    const float* ea     = (const float*)d_in[2];
    const float* w_sg1_1 = (const float*)d_in[4];
    const float* w_sg1_2 = (const float*)d_in[5];
    const float* w_sg1_3 = (const float*)d_in[6];
    const float* w_sg2_1 = (const float*)d_in[7];
    const float* w_sg2_2 = (const float*)d_in[8];
    const float* w_sg2_3 = (const float*)d_in[9];
    const float* w_sg3_1 = (const float*)d_in[10];
    const float* w_sg3_2 = (const float*)d_in[11];
    const float* w_sg3_3 = (const float*)d_in[12];
    const float* w_dg1_1 = (const float*)d_in[13];
    const float* w_dg1_2 = (const float*)d_in[14];
    const float* w_dg2_1 = (const float*)d_in[15];
    const float* w_dg2_2 = (const float*)d_in[16];
    const float* w_f1    = (const float*)d_in[17];
    const float* w_f2_1  = (const float*)d_in[18];
    const float* w_f2_2  = (const float*)d_in[19];
    const float* w_f2_3  = (const float*)d_in[20];

    // workspace bump allocator
    char* ws = (char*)d_ws;
    size_t off = 0;
    auto alloc = [&](size_t bytes) -> char* {
        char* p = ws + off;
        off += (bytes + 255) & ~(size_t)255;
        return p;
    };
    const int BN_CHUNKS_MAX = (EDGES_DYN + BN_RPC - 1) / BN_RPC;  // 68
    float*    bufA = (float*)alloc((size_t)EDGES_DYN * 256 * 4); // dyn msg / h2
    float*    bufB = (float*)alloc((size_t)EDGES_DYN * 192 * 4); // dyn h1
    unsigned* agg  = (unsigned*)alloc((size_t)N * 320 * 4);
    float*    hA   = (float*)alloc((size_t)N * 256 * 4);
    float*    hB   = (float*)alloc((size_t)N * 256 * 4);
    float*    sg1  = (float*)alloc((size_t)N * 64 * 4);
    float*    sg2  = (float*)alloc((size_t)N * 128 * 4);
    float*    sg3  = (float*)alloc((size_t)N * 192 * 4);
    float*    dg1  = (float*)alloc((size_t)N * 128 * 4);
    float*    dg2  = (float*)alloc((size_t)N * 192 * 4);
    float*    xnb  = (float*)alloc((size_t)N * 128 * 4);
    int*      knn  = (int*)alloc((size_t)N * KNN_K * 4);
    float*    cat2 = (float*)alloc((size_t)N * 1728 * 4);       // [f1 | cat1]
    float*    h512 = (float*)alloc((size_t)N * 512 * 4);
    float*    h256 = (float*)alloc((size_t)N * 256 * 4);
    float*    mean = (float*)alloc(1024 * 4);
    float*    rstd = (float*)alloc(1024 * 4);
    float*    psum  = (float*)alloc((size_t)BN_CHUNKS_MAX * 1024 * 4);
    float*    psum2 = (float*)alloc((size_t)BN_CHUNKS_MAX * 1024 * 4);

    auto gemm = [&](const float* A, const float* B, float* C,
                    int M, int K, int Nc, int lda, int ldb, int ldc) {
        dim3 g(Nc / 32, (M + 255) / 256), b(128);
        gemm_f32_wmma<<<g, b, 0, stream>>>(A, B, C, M, K, Nc, lda, ldb, ldc);
    };
    auto bn = [&](float* Xp, int M, int C, int ld, float slope) {
        const int chunks = (M + BN_RPC - 1) / BN_RPC;
        bn_partial<<<dim3(C / 32, chunks), dim3(32, 8), 0, stream>>>(
            Xp, M, C, ld, psum, psum2);
        bn_finalize<<<dim3((C + 255) / 256), dim3(256), 0, stream>>>(
            psum, psum2, chunks, C, M, mean, rstd);
        bn_lrelu<<<dim3(ew_blocks((size_t)M * C)), dim3(256), 0, stream>>>(
            Xp, mean, rstd, slope, M, C, ld);
    };
    auto static_conv = [&](const float* xin, int d,
                           const float* w1, const float* w2, const float* w3,
                           int c1, int c2, int c3, float* outp) {
        const int Cw = 2 * d + DE_DIM;
        hipMemsetAsync(agg, 0, (size_t)N * Cw * 4, stream);
        static_scatter<<<dim3(8192), dim3(256), 0, stream>>>(xin, ei, ea, agg, d);
        decode_agg<<<dim3(4096), dim3(256), 0, stream>>>(agg, (size_t)N * Cw);
        gemm((const float*)agg, w1, hA, N, Cw, c1, Cw, c1, c1);  bn(hA, N, c1, c1, 0.1f);
        gemm(hA, w2, hB, N, c1, c2, c1, c2, c2);                 bn(hB, N, c2, c2, 0.1f);
        gemm(hB, w3, outp, N, c2, c3, c2, c3, c3);               bn(outp, N, c3, c3, 0.1f);
    };
    auto dynamic_conv = [&](const float* xin, int d,
                            const float* w1, const float* w2,
                            int c1, int c2, float* outp) {
        rownorm<<<dim3(N), dim3(128), 0, stream>>>(xin, xnb, d);
        knn_kernel<<<dim3(N / 4), dim3(128), 0, stream>>>(xnb, knn, d);
        dyn_msg<<<dim3(8192), dim3(256), 0, stream>>>(xin, knn, bufA, d);
        const int Me = EDGES_DYN;
        gemm(bufA, w1, bufB, Me, 2 * d, c1, 2 * d, c1, c1);  bn(bufB, Me, c1, c1, 0.2f);
        gemm(bufB, w2, bufA, Me, c1, c2, c1, c2, c2);        bn(bufA, Me, c2, c2, 0.2f);
        dyn_segmax<<<dim3(4096), dim3(256), 0, stream>>>(bufA, outp, c2);
    };

    // ---- graph conv stack -------------------------------------------------
    static_conv(x,   64,  w_sg1_1, w_sg1_2, w_sg1_3, 128,  64,  64, sg1);
    static_conv(sg1, 64,  w_sg2_1, w_sg2_2, w_sg2_3, 160, 128, 128, sg2);
    static_conv(sg2, 128, w_sg3_1, w_sg3_2, w_sg3_3, 256, 192, 192, sg3);
    dynamic_conv(sg1, 64,  w_dg1_1, w_dg1_2, 128, 128, dg1);
    dynamic_conv(dg1, 128, w_dg2_1, w_dg2_2, 192, 192, dg2);

    // ---- cat1 into cat2 columns [1024..1727] ------------------------------
    float* cat1 = cat2 + 1024;
    copy_cols<<<dim3(ew_blocks((size_t)N *  64)), dim3(256), 0, stream>>>(sg1,  64, cat1 + 0,   1728);
    copy_cols<<<dim3(ew_blocks((size_t)N * 128)), dim3(256), 0, stream>>>(dg1, 128, cat1 + 64,  1728);
    copy_cols<<<dim3(ew_blocks((size_t)N * 192)), dim3(256), 0, stream>>>(dg2, 192, cat1 + 192, 1728);
    copy_cols<<<dim3(ew_blocks((size_t)N * 128)), dim3(256), 0, stream>>>(sg2, 128, cat1 + 384, 1728);
    copy_cols<<<dim3(ew_blocks((size_t)N * 192)), dim3(256), 0, stream>>>(sg3, 192, cat1 + 512, 1728);

    // ---- head: f1 written into cat2 cols [0..1023] (strided views) --------
    gemm(cat1, w_f1, cat2, N, 704, 1024, 1728, 1024, 1728);  bn(cat2, N, 1024, 1728, 0.2f);
    gemm(cat2, w_f2_1, h512, N, 1728, 512, 1728, 512, 512);  bn(h512, N, 512, 512, 0.2f);
    gemm(h512, w_f2_2, h256, N, 512, 256, 512, 256, 256);    bn(h256, N, 256, 256, 0.2f);
    gemm(h256, w_f2_3, (float*)d_out, N, 256, 128, 256, 128, 128);
}